// RelativeMultiHeadAttention_17721035063725
// MI455X (gfx1250) — compile-verified
//
#include <hip/hip_runtime.h>

typedef float v2f __attribute__((ext_vector_type(2)));
typedef float v8f __attribute__((ext_vector_type(8)));

#define S_TOT   1025
#define S_LEN   1024
#define DIM     1024
#define NH      16
#define DH      64
#define PE_ROWS 2047   // 2*S_LEN - 1

__device__ __forceinline__ v8f wmma_f32(v2f a, v2f b, v8f c) {
  // V_WMMA_F32_16X16X4_F32 : D = A(16x4) * B(4x16) + C(16x16)
  return __builtin_amdgcn_wmma_f32_16x16x4_f32(
      /*neg_a=*/false, a, /*neg_b=*/false, b,
      /*c_mod=*/(short)0, c, /*reuse_a=*/false, /*reuse_b=*/false);
}

// ---------------------------------------------------------------------------
// Sinusoidal positional embedding: pe[r, c], r in [0,2047), t = r - 1023.
// emb[:, 2i] = sin(t / 10000^(2i/d)), emb[:, 2i+1] = cos(...), scaled d^-0.5.
// ---------------------------------------------------------------------------
__global__ void pe_gen_kernel(float* __restrict__ pe) {
  int idx = blockIdx.x * blockDim.x + threadIdx.x;
  if (idx >= PE_ROWS * DIM) return;
  int row = idx >> 10;
  int c   = idx & 1023;
  int i2  = c & ~1;  // 2*(c/2)
  float t   = (float)(row - (S_LEN - 1));
  float ang = t * expf(-9.210340371976184f * (float)i2 * (1.0f / (float)DIM));
  float v   = (c & 1) ? cosf(ang) : sinf(ang);
  pe[idx] = v * 0.03125f;   // 1024^-0.5
}

// ---------------------------------------------------------------------------
// Generic C = A @ B^T GEMM. A:[M,K] row-major, B:[N,K] row-major, C:[M,N].
// Register-blocked: each wave computes a 32x64 tile (2 A-frags x 4 B-frags
// -> 8 V_WMMA_F32_16X16X4_F32 per 6 b64 loads). Block = 4 waves = 32x256.
// grid = (ceil(M/32), N/256). Requires N % 256 == 0, K % 16 == 0.
// ---------------------------------------------------------------------------
__global__ void __launch_bounds__(128)
gemm_nt_kernel(const float* __restrict__ A, const float* __restrict__ B,
               float* __restrict__ C, int M, int N, int K) {
  const int lane = threadIdx.x & 31;
  const int wave = threadIdx.x >> 5;
  const int half = lane >> 4;
  const int r    = lane & 15;
  const int m0 = blockIdx.x * 32;
  const int n0 = blockIdx.y * 256 + wave * 64;
  if (m0 >= M || n0 >= N) return;

  int ar0 = m0 + r;      if (ar0 > M - 1) ar0 = M - 1;   // clamp ragged M tile
  int ar1 = m0 + 16 + r; if (ar1 > M - 1) ar1 = M - 1;
  const float* ap0 = A + (size_t)ar0 * K + 2 * half;
  const float* ap1 = A + (size_t)ar1 * K + 2 * half;
  const float* bp0 = B + (size_t)(n0 + r) * K + 2 * half;
  const float* bp1 = B + (size_t)(n0 + 16 + r) * K + 2 * half;
  const float* bp2 = B + (size_t)(n0 + 32 + r) * K + 2 * half;
  const float* bp3 = B + (size_t)(n0 + 48 + r) * K + 2 * half;

  const v8f vzero = {0.f, 0.f, 0.f, 0.f, 0.f, 0.f, 0.f, 0.f};
  v8f acc[2][4];
#pragma unroll
  for (int mi = 0; mi < 2; ++mi)
#pragma unroll
    for (int n = 0; n < 4; ++n) acc[mi][n] = vzero;

  for (int kb = 0; kb < K; kb += 16) {
    // stream the next 1KB of each row into the cache hierarchy; locality=3
    // -> WGP-scope prefetch (fills all cache levels). Speculative prefetch is
    // architecturally safe past the end of the buffer.
    __builtin_prefetch(ap0 + kb + 256, 0, 3);
    __builtin_prefetch(ap1 + kb + 256, 0, 3);
    __builtin_prefetch(bp0 + kb + 256, 0, 3);
    __builtin_prefetch(bp1 + kb + 256, 0, 3);
    __builtin_prefetch(bp2 + kb + 256, 0, 3);
    __builtin_prefetch(bp3 + kb + 256, 0, 3);
#pragma unroll
    for (int ki = 0; ki < 16; ki += 4) {
      const int k = kb + ki;
      v2f a0, a1, b0, b1, b2, b3;
      a0.x = ap0[k]; a0.y = ap0[k + 1];
      a1.x = ap1[k]; a1.y = ap1[k + 1];
      b0.x = bp0[k]; b0.y = bp0[k + 1];
      b1.x = bp1[k]; b1.y = bp1[k + 1];
      b2.x = bp2[k]; b2.y = bp2[k + 1];
      b3.x = bp3[k]; b3.y = bp3[k + 1];
      acc[0][0] = wmma_f32(a0, b0, acc[0][0]);
      acc[0][1] = wmma_f32(a0, b1, acc[0][1]);
      acc[0][2] = wmma_f32(a0, b2, acc[0][2]);
      acc[0][3] = wmma_f32(a0, b3, acc[0][3]);
      acc[1][0] = wmma_f32(a1, b0, acc[1][0]);
      acc[1][1] = wmma_f32(a1, b1, acc[1][1]);
      acc[1][2] = wmma_f32(a1, b2, acc[1][2]);
      acc[1][3] = wmma_f32(a1, b3, acc[1][3]);
    }
  }

#pragma unroll
  for (int mi = 0; mi < 2; ++mi) {
#pragma unroll
    for (int r2 = 0; r2 < 8; ++r2) {
      int row = m0 + mi * 16 + r2 + 8 * half;  // C frag: VGPR r2 -> rows r2/r2+8
      if (row < M) {
#pragma unroll
        for (int n = 0; n < 4; ++n)
          C[(size_t)row * N + n0 + n * 16 + r] = acc[mi][n][r2];
      }
    }
  }
}

// ---------------------------------------------------------------------------
// Flash-style relative attention. One wave (32 threads) per (qtile, head, b).
// S[qi,kj] = (q+cb)·k  +  [qi>=1 && kj>=1] * (q+pb)·pe_key[kj-qi+1023]
// scaled by dh^-0.5 = 0.125, causal + src_mask, online softmax, out = P@V.
// qkv layout: [b*1025, 3*1024] with col = c*1024 + head*64 + dd.
// ---------------------------------------------------------------------------
__global__ void __launch_bounds__(32)
rel_attn_kernel(const float* __restrict__ qkv, const float* __restrict__ pe_key,
                const unsigned char* __restrict__ src_mask,
                const float* __restrict__ cbias, const float* __restrict__ pbias,
                float* __restrict__ out) {
  const int lane = threadIdx.x & 31;
  const int half = lane >> 4;
  const int r    = lane & 15;
  const int qt   = blockIdx.x;
  const int head = blockIdx.y;
  const int b    = blockIdx.z;
  const int q0   = qt * 16;

  __shared__ float ldsT[16][34];   // Toeplitz band tile T = Qp @ PE^T (16x32)
  __shared__ float ldsP[16][20];   // probability tile for P@V relayout

  const v8f vzero = {0.f, 0.f, 0.f, 0.f, 0.f, 0.f, 0.f, 0.f};
  float mrow[8], lrow[8];
  v8f oacc[4];
#pragma unroll
  for (int i = 0; i < 8; ++i) { mrow[i] = -1e30f; lrow[i] = 0.f; }
#pragma unroll
  for (int n = 0; n < 4; ++n) oacc[n] = vzero;

  const size_t bbase = (size_t)b * S_TOT;
  int qrow = q0 + r; if (qrow > S_TOT - 1) qrow = S_TOT - 1;
  const float* qp = qkv + (bbase + qrow) * (3 * DIM) + head * DH + 2 * half;
  const float* cb = cbias + head * DH + 2 * half;
  const float* pb = pbias + head * DH + 2 * half;

  for (int kt = 0; kt <= qt; ++kt) {
    const int k0 = kt * 16;
    int krow = k0 + r; if (krow > S_TOT - 1) krow = S_TOT - 1;
    const float* kp = qkv + (bbase + krow) * (3 * DIM) + DIM + head * DH + 2 * half;

    // pe band: l = (kj - qi) + 1023 = base + (j' - i' + 15), base = k0-q0+1008
    const int basepe = k0 - q0 + 1008;
    int l0 = basepe + r;      if (l0 < 0) l0 = 0; if (l0 > PE_ROWS - 1) l0 = PE_ROWS - 1;
    int l1 = basepe + 16 + r; if (l1 < 0) l1 = 0; if (l1 > PE_ROWS - 1) l1 = PE_ROWS - 1;
    const float* pe0 = pe_key + (size_t)l0 * DIM + head * DH + 2 * half;
    const float* pe1 = pe_key + (size_t)l1 * DIM + head * DH + 2 * half;

    v8f sc = vzero, t0 = vzero, t1 = vzero;
#pragma unroll 4
    for (int kk = 0; kk < DH; kk += 4) {
      float qa = qp[kk], qb = qp[kk + 1];
      v2f ac;   ac.x   = qa + cb[kk]; ac.y   = qb + cb[kk + 1];
      v2f apos; apos.x = qa + pb[kk]; apos.y = qb + pb[kk + 1];
      v2f bk; bk.x = kp[kk];  bk.y = kp[kk + 1];
      v2f b0; b0.x = pe0[kk]; b0.y = pe0[kk + 1];
      v2f b1; b1.x = pe1[kk]; b1.y = pe1[kk + 1];
      sc = wmma_f32(ac,   bk, sc);
      t0 = wmma_f32(apos, b0, t0);
      t1 = wmma_f32(apos, b1, t1);
    }

#pragma unroll
    for (int r2 = 0; r2 < 8; ++r2) {
      int ip = r2 + 8 * half;
      ldsT[ip][r]      = t0[r2];
      ldsT[ip][16 + r] = t1[r2];
    }
    __syncthreads();

    // masks + online softmax update (row = r2 + 8*half, 16 cols across lanes)
#pragma unroll
    for (int r2 = 0; r2 < 8; ++r2) {
      int ip = r2 + 8 * half;
      int qi = q0 + ip;
      int kj = k0 + r;
      float sv = sc[r2];
      if (qi >= 1 && kj >= 1) sv += ldsT[ip][r - ip + 15];  // in [0,30]
      sv *= 0.125f;  // dh^-0.5
      bool masked = (kj > qi) || (qi >= S_TOT) || (kj >= S_TOT) ||
                    (src_mask[bbase + kj] != 0);
      if (masked) sv = -1e30f;

      float mx = sv;
#pragma unroll
      for (int off = 1; off < 16; off <<= 1)
        mx = fmaxf(mx, __shfl_xor(mx, off, 32));
      float mnew  = fmaxf(mrow[r2], mx);
      float alpha = __expf(mrow[r2] - mnew);
      mrow[r2] = mnew;
      float pv = __expf(sv - mnew);
      float ps = pv;
#pragma unroll
      for (int off = 1; off < 16; off <<= 1)
        ps += __shfl_xor(ps, off, 32);
      lrow[r2] = lrow[r2] * alpha + ps;
#pragma unroll
      for (int n = 0; n < 4; ++n) oacc[n][r2] = oacc[n][r2] * alpha;
      ldsP[ip][r] = pv;
    }
    __syncthreads();

    // O += P(16x16) @ V(16x64): A-frag from LDS, B-frag = rows of V
#pragma unroll
    for (int kk = 0; kk < 16; kk += 4) {
      v2f ap2;
      ap2.x = ldsP[r][kk + 2 * half];
      ap2.y = ldsP[r][kk + 2 * half + 1];
      int v0 = k0 + kk + 2 * half;     if (v0 > S_TOT - 1) v0 = S_TOT - 1;
      int v1 = k0 + kk + 2 * half + 1; if (v1 > S_TOT - 1) v1 = S_TOT - 1;
      const float* vp0 = qkv + (bbase + v0) * (3 * DIM) + 2 * DIM + head * DH;
      const float* vp1 = qkv + (bbase + v1) * (3 * DIM) + 2 * DIM + head * DH;
#pragma unroll
      for (int n = 0; n < 4; ++n) {
        v2f bv; bv.x = vp0[n * 16 + r]; bv.y = vp1[n * 16 + r];
        oacc[n] = wmma_f32(ap2, bv, oacc[n]);
      }
    }
    __syncthreads();
  }

  // epilogue: divide by row sums, write [b, s, h*dh]
#pragma unroll
  for (int r2 = 0; r2 < 8; ++r2) {
    int ip = r2 + 8 * half;
    int qi = q0 + ip;
    if (qi < S_TOT) {
      float inv = 1.0f / lrow[r2];
#pragma unroll
      for (int n = 0; n < 4; ++n)
        out[(bbase + qi) * DIM + head * DH + n * 16 + r] = oacc[n][r2] * inv;
    }
  }
}

// ---------------------------------------------------------------------------
extern "C" void kernel_launch(void* const* d_in, const int* in_sizes, int n_in,
                              void* d_out, int out_size, void* d_ws, size_t ws_size,
                              hipStream_t stream) {
  const float*         x     = (const float*)d_in[0];         // [2,1025,1024]
  const unsigned char* smask = (const unsigned char*)d_in[1]; // [2,1025] bool
  const float*         W_qkv = (const float*)d_in[2];         // [3072,1024]
  const float*         W_pe  = (const float*)d_in[3];         // [1024,1024]
  const float*         W_o   = (const float*)d_in[4];         // [1024,1024]
  const float*         cb    = (const float*)d_in[5];         // [16,1,64]
  const float*         pb    = (const float*)d_in[6];         // [16,1,64]
  float*               outp  = (float*)d_out;                 // [2,1025,1024]

  const int ROWS = 2 * S_TOT;  // 2050
  float* ws       = (float*)d_ws;
  float* qkv_buf  = ws;                                   // 2050*3072
  float* pe_buf   = qkv_buf + (size_t)ROWS * 3 * DIM;     // 2047*1024
  float* pekey    = pe_buf + (size_t)PE_ROWS * DIM;       // 2047*1024
  float* attn_out = pekey + (size_t)PE_ROWS * DIM;        // 2050*1024

  // 1) sinusoidal pe
  {
    int n = PE_ROWS * DIM;
    pe_gen_kernel<<<(n + 255) / 256, 256, 0, stream>>>(pe_buf);
  }
  // 2) qkv = x @ W_qkv^T  : [2050,1024] x [3072,1024]^T
  gemm_nt_kernel<<<dim3((ROWS + 31) / 32, (3 * DIM) / 256), 128, 0, stream>>>(
      x, W_qkv, qkv_buf, ROWS, 3 * DIM, DIM);
  // 3) pe_key = pe @ W_pe^T : [2047,1024] x [1024,1024]^T
  gemm_nt_kernel<<<dim3((PE_ROWS + 31) / 32, DIM / 256), 128, 0, stream>>>(
      pe_buf, W_pe, pekey, PE_ROWS, DIM, DIM);
  // 4) flash relative attention
  rel_attn_kernel<<<dim3((S_TOT + 15) / 16, NH, 2), 32, 0, stream>>>(
      qkv_buf, pekey, smask, cb, pb, attn_out);
  // 5) out = attn_out @ W_o^T
  gemm_nt_kernel<<<dim3((ROWS + 31) / 32, DIM / 256), 128, 0, stream>>>(
      attn_out, W_o, outp, ROWS, DIM, DIM);
}